// ConcatAttentionFusion_51256139710853
// MI455X (gfx1250) — compile-verified
//
#include <hip/hip_runtime.h>
#include <math.h>

// Fused concat + self-attention (flash-attention style, fp32 WMMA + TDM).
//
// Roofline: 103 GFLOP fp32 vs ~100MB HBM traffic -> compute bound on the
// matrix path. Unscaled logits (std ~ sqrt(768) = 27.7) make bf16 QK^T
// numerically unusable, so everything runs through V_WMMA_F32_16X16X4_F32.
// Online softmax avoids materializing the 128MB similarity tensor.
// K/V tiles are double-buffered in LDS and fed by the Tensor Data Mover
// (tensor_load_to_lds, TENSORcnt) so the DMA overlaps the WMMA stream.

typedef __attribute__((ext_vector_type(2))) float v2f;
typedef __attribute__((ext_vector_type(8))) float v8f;
typedef __attribute__((ext_vector_type(4))) unsigned int u32x4;
typedef __attribute__((ext_vector_type(8))) int i32x8;
typedef __attribute__((ext_vector_type(4))) int i32x4;

#define BDIM 768      // embedding dim
#define SEQ  1024     // per-source sequence length
#define NTOT 2048     // concat sequence length
#define QROWS 32      // query rows per block (2 waves x 16)
#define KT   16       // key/value tile rows
#define NKT  (NTOT / KT)
#define LSTR 772      // padded LDS row stride (floats): kills bank conflicts, keeps 16B align
#define PSTR 18       // P-tile row stride (floats), keeps 8B align for v2f

#if defined(__has_builtin)
#if __has_builtin(__builtin_amdgcn_tensor_load_to_lds)
#define HAVE_TDM 1
#endif
#endif
#if __has_include(<hip/amd_detail/amd_gfx1250_TDM.h>)
#define TDM_6ARG 1   // therock headers -> 6-arg builtin arity
#endif

__device__ __forceinline__ v8f wmma16x16x4(v2f a, v2f b, v8f c) {
  // emits v_wmma_f32_16x16x4_f32
  return __builtin_amdgcn_wmma_f32_16x16x4_f32(false, a, false, b, (short)0, c,
                                               false, false);
}

__device__ __forceinline__ const float* rowbase(const float* g, const float* l,
                                                int b, int n) {
  return (n < SEQ) ? g + ((size_t)b * SEQ + n) * BDIM
                   : l + ((size_t)b * SEQ + (n - SEQ)) * BDIM;
}

// TDM: DMA `nrows` rows of 768 f32 from global into LDS with padded row
// stride LSTR, via iterate mode (per-iteration tile = 768x1, nrows iters).
__device__ __forceinline__ void tdm_load_rows(unsigned lds_byte_off,
                                              const float* gsrc, int nrows) {
#ifdef HAVE_TDM
  const unsigned long long ga = (unsigned long long)(uintptr_t)gsrc;

  u32x4 g0;
  g0[0] = 1u;                                   // count=1 (valid), user mode
  g0[1] = lds_byte_off;                         // lds_addr (bytes)
  g0[2] = (unsigned)ga;                         // global_addr[31:0]
  g0[3] = (unsigned)((ga >> 32) & 0x01FFFFFFu)  // global_addr[56:32]
          | (2u << 30);                         // type=2 ("image")

  i32x8 g1;
  g1[0] = (2 << 16) | (1 << 19);   // data_size=4B, iterate_enable=1
  g1[1] = (int)(768u << 16);       // tensor_dim0[15:0] = 768 (bits 63:48)
  g1[2] = (int)(1u << 16);         // tensor_dim0 hi=0 | tensor_dim1 lo = 1
  g1[3] = (int)(768u << 16);       // tensor_dim1 hi=0 | tile_dim0 = 768
  g1[4] = 1;                       // tile_dim1 = 1, tile_dim2 = 0
  g1[5] = 768;                     // tensor_dim0_stride[31:0]
  g1[6] = 0;                       // stride0 hi | stride1 lo
  g1[7] = 0;                       // stride1 hi

  i32x4 g2;
  g2[0] = 1;                       // tensor_dim2 (unused)
  g2[1] = LSTR;                    // lds_addr_increment per iteration (elems)
  g2[2] = 768;                     // global_addr_increment per iteration (elems)
  g2[3] = (nrows - 1) << 16;       // iterate_count (0 => 1x)

  i32x4 g3 = {0, 0, 0, 0};

#ifdef TDM_6ARG
  i32x8 gz = {0, 0, 0, 0, 0, 0, 0, 0};
  __builtin_amdgcn_tensor_load_to_lds(g0, g1, g2, g3, gz, 0);
#else
  __builtin_amdgcn_tensor_load_to_lds(g0, g1, g2, g3, 0);
#endif
#else
  (void)lds_byte_off; (void)gsrc; (void)nrows;
#endif
}

__global__ __launch_bounds__(64, 1)
void fused_concat_attn(const float* __restrict__ gemb,
                       const float* __restrict__ lemb,
                       float* __restrict__ out) {
  __shared__ float Qs[QROWS][LSTR];     // 32 query rows x 768 (padded)
  __shared__ float Ks[2][KT][LSTR];     // double-buffered K/V tile (K == V)
  __shared__ float Ps[2][16][PSTR];     // per-wave 16x16 S/P staging
  __shared__ float rowM[2][16];         // running row max
  __shared__ float rowL[2][16];         // running row sum
  __shared__ float rowS[2][16];         // per-iteration rescale factor

  const int tid  = threadIdx.x;
  const int w    = tid >> 5;            // wave id (0..1)
  const int lane = tid & 31;
  const int ln   = lane & 15;           // lane within half-wave
  const int hi   = lane >> 4;           // half-wave select

  const int b  = blockIdx.x / (NTOT / QROWS);
  const int q0 = (blockIdx.x % (NTOT / QROWS)) * QROWS;

  // ---- kick off Q tile + first K tile DMAs (wave 0 drives the TDM) ----
#ifdef HAVE_TDM
  if (w == 0) {
    tdm_load_rows((unsigned)(uintptr_t)&Qs[0][0],
                  rowbase(gemb, lemb, b, q0), QROWS);
    tdm_load_rows((unsigned)(uintptr_t)&Ks[0][0][0],
                  rowbase(gemb, lemb, b, 0), KT);
  }
#else
  {
    const float* qb = rowbase(gemb, lemb, b, q0);
    for (int i = tid; i < QROWS * (BDIM / 4); i += 64) {
      const int r = i / (BDIM / 4);
      const int c = (i % (BDIM / 4)) * 4;
      *(float4*)&Qs[r][c] = *(const float4*)(qb + (size_t)r * BDIM + c);
    }
    const float* kb = rowbase(gemb, lemb, b, 0);
    for (int i = tid; i < KT * (BDIM / 4); i += 64) {
      const int r = i / (BDIM / 4);
      const int c = (i % (BDIM / 4)) * 4;
      *(float4*)&Ks[0][r][c] = *(const float4*)(kb + (size_t)r * BDIM + c);
    }
  }
#endif

  if (lane < 16) {
    rowM[w][lane] = -INFINITY;
    rowL[w][lane] = 0.0f;
  }

  // per-wave output accumulator: 16 rows x 768 cols = 48 tiles of 16x16
  v8f O[BDIM / 16];
#pragma unroll
  for (int t = 0; t < BDIM / 16; ++t)
#pragma unroll
    for (int r = 0; r < 8; ++r) O[t][r] = 0.0f;

  for (int kt = 0; kt < NKT; ++kt) {
    const int cur = kt & 1;
    const int nxt = cur ^ 1;

    __syncthreads();  // all waves done reading Ks[nxt] from iteration kt-1
#ifdef HAVE_TDM
    if (w == 0) {
      if (kt + 1 < NKT) {
        tdm_load_rows((unsigned)(uintptr_t)&Ks[nxt][0][0],
                      rowbase(gemb, lemb, b, (kt + 1) * KT), KT);
        __builtin_amdgcn_s_wait_tensorcnt((short)1);  // tile kt has landed
      } else {
        __builtin_amdgcn_s_wait_tensorcnt((short)0);
      }
    }
#else
    if (kt + 1 < NKT) {
      const float* kb = rowbase(gemb, lemb, b, (kt + 1) * KT);
      for (int i = tid; i < KT * (BDIM / 4); i += 64) {
        const int r = i / (BDIM / 4);
        const int c = (i % (BDIM / 4)) * 4;
        *(float4*)&Ks[nxt][r][c] = *(const float4*)(kb + (size_t)r * BDIM + c);
      }
    }
#endif
    __syncthreads();  // publish tile kt to both waves

    // ---- S(16x16) = Qtile(16x768) . Ktile(16x768)^T via 192 fp32 WMMAs ----
    v8f s = {};
    {
      // A frag: lane reads Qs[ln][d + 2*hi .. +1]; B frag: Ks[ln][d + 2*hi .. +1]
      const float* qrow = &Qs[w * 16 + ln][2 * hi];
      const float* krow = &Ks[cur][ln][2 * hi];
#pragma unroll 8
      for (int d = 0; d < BDIM; d += 4) {
        const v2f a  = *(const v2f*)(qrow + d);
        const v2f bb = *(const v2f*)(krow + d);
        s = wmma16x16x4(a, bb, s);
      }
    }

    // stage S tile to LDS (C layout: VGPR r -> row r + 8*hi, col ln)
#pragma unroll
    for (int r = 0; r < 8; ++r) Ps[w][r + 8 * hi][ln] = s[r];
    __syncthreads();

    // ---- online softmax: one lane per row ----
    if (lane < 16) {
      const float mold = rowM[w][lane];
      float tmax = Ps[w][lane][0];
#pragma unroll
      for (int j = 1; j < 16; ++j) tmax = fmaxf(tmax, Ps[w][lane][j]);
      const float mnew  = fmaxf(mold, tmax);
      const float scale = __expf(mold - mnew);  // == 1.0 exactly when max unchanged
      float sum = 0.0f;
#pragma unroll
      for (int j = 0; j < 16; ++j) {
        const float p = __expf(Ps[w][lane][j] - mnew);
        Ps[w][lane][j] = p;
        sum += p;
      }
      rowL[w][lane] = rowL[w][lane] * scale + sum;
      rowM[w][lane] = mnew;
      rowS[w][lane] = scale;
    }
    __syncthreads();

    // ---- rescale accumulator only when some row max actually moved ----
    {
      float sc[8];
      bool need = false;
#pragma unroll
      for (int r = 0; r < 8; ++r) {
        sc[r] = rowS[w][r + 8 * hi];
        need = need || (sc[r] != 1.0f);
      }
      if (__builtin_amdgcn_ballot_w32(need)) {  // wave-uniform branch
#pragma unroll
        for (int t = 0; t < BDIM / 16; ++t)
#pragma unroll
          for (int r = 0; r < 8; ++r) O[t][r] *= sc[r];
      }
    }

    // ---- O(16x768) += P(16x16) . V(16x768) ----
    v2f pa[4];  // P A-frags for the four K=4 chunks (reused by all 48 tiles)
#pragma unroll
    for (int kc = 0; kc < 4; ++kc)
      pa[kc] = *(const v2f*)&Ps[w][ln][4 * kc + 2 * hi];

#pragma unroll
    for (int t = 0; t < BDIM / 16; ++t) {
      const int c0 = 16 * t;
#pragma unroll
      for (int kc = 0; kc < 4; ++kc) {
        const int kk = 4 * kc + 2 * hi;
        v2f bb;
        bb.x = Ks[cur][kk][c0 + ln];      // B row K = kk,     col N = ln
        bb.y = Ks[cur][kk + 1][c0 + ln];  // B row K = kk + 1, col N = ln
        O[t] = wmma16x16x4(pa[kc], bb, O[t]);
      }
    }
  }

  // ---- epilogue: divide by row sums, coalesced stores ----
  if (lane < 16) rowL[w][lane] = 1.0f / rowL[w][lane];
  __syncthreads();

  float inv[8];
#pragma unroll
  for (int r = 0; r < 8; ++r) inv[r] = rowL[w][r + 8 * hi];

  float* obase = out + ((size_t)b * NTOT + q0 + w * 16) * BDIM;
#pragma unroll
  for (int r = 0; r < 8; ++r) {
    float* orow = obase + (size_t)(r + 8 * hi) * BDIM + ln;
#pragma unroll
    for (int t = 0; t < BDIM / 16; ++t)
      orow[16 * t] = O[t][r] * inv[r];
  }
}

extern "C" void kernel_launch(void* const* d_in, const int* in_sizes, int n_in,
                              void* d_out, int out_size, void* d_ws, size_t ws_size,
                              hipStream_t stream) {
  const float* gemb = (const float*)d_in[0];
  const float* lemb = (const float*)d_in[1];
  float* out = (float*)d_out;

  const int B = in_sizes[0] / (SEQ * BDIM);  // 8 for the reference shapes
  dim3 grid(B * (NTOT / QROWS));             // 8 * 64 = 512 blocks
  dim3 block(64);                            // 2 waves (wave32)
  hipLaunchKernelGGL(fused_concat_attn, grid, block, 0, stream, gemb, lemb, out);
}